// TACTiS_55619826483602
// MI455X (gfx1250) — compile-verified
//
#include <hip/hip_runtime.h>
#include <hip/hip_bf16.h>

// TACTiS-style transformer + DSF flow for MI455X (gfx1250), fp32 WMMA path.
// All GEMMs (projections, attention, FF) run on V_WMMA_F32_16X16X4_F32.
// v2: 32x64 C-tile per wave (B fragments reused 2x), 32-bit offset addressing
//     so global loads take the saddr+voffset form instead of per-lane 64-bit
//     pointer chains.

typedef __attribute__((ext_vector_type(2))) float v2f;
typedef __attribute__((ext_vector_type(8))) float v8f;

#define EPSF 1e-06f

__device__ __forceinline__ v8f wmma_f32(v2f a, v2f b, v8f c) {
  // D = A(16x4) * B(4x16) + C(16x16), fp32
  return __builtin_amdgcn_wmma_f32_16x16x4_f32(false, a, false, b, (short)0, c,
                                               false, false);
}

// ---------------------------------------------------------------------------
// WMMA GEMM: Out[M,N] = act(A[M,K] @ W[K,N] + bias). 128x64 per block,
// 32x64 per wave (8 accumulators, double A fragments -> B reuse x2).
// gridDim.z batches over heads via wz/bz/oz strides. M%128==0, N%64==0, K%4==0.
// ---------------------------------------------------------------------------
__global__ __launch_bounds__(128) void wmma_gemm_kernel(
    const float* __restrict__ A, int lda,
    const float* __restrict__ W, int ldw, int wz,
    const float* __restrict__ bias, int bz,
    float* __restrict__ Out, int ldo, int oz,
    int K, int relu)
{
  W    += (unsigned)(blockIdx.z * wz);
  bias += (unsigned)(blockIdx.z * bz);
  Out  += (unsigned)(blockIdx.z * oz);
  const int lane = threadIdx.x & 31;
  const int wave = threadIdx.x >> 5;
  const int n    = lane & 15;            // col within 16-tile / A row (m)
  const int kb   = (lane >> 4) << 1;     // K sub-offset: 0 or 2
  const int hi8  = (lane >> 4) << 3;     // C-row offset: 0 or 8
  const int row0 = ((blockIdx.y << 2) | wave) << 5;   // 32 rows per wave
  const int col0 = blockIdx.x << 6;                   // 64 cols per block

  v8f c00 = {}, c01 = {}, c02 = {}, c03 = {};   // rows [row0, row0+16)
  v8f c10 = {}, c11 = {}, c12 = {}, c13 = {};   // rows [row0+16, row0+32)

  const unsigned aoff  = (unsigned)(row0 + n) * (unsigned)lda + (unsigned)kb;
  const unsigned astep = 16u * (unsigned)lda;
  const unsigned wbase = (unsigned)kb * (unsigned)ldw + (unsigned)(col0 + n);

  for (int k = 0; k < K; k += 4) {
    v2f a0 = *(const v2f*)(A + aoff + (unsigned)k);
    v2f a1 = *(const v2f*)(A + aoff + astep + (unsigned)k);
    const unsigned wo = wbase + (unsigned)k * (unsigned)ldw;
    v2f b0 = { W[wo +  0], W[wo + (unsigned)ldw +  0] };
    v2f b1 = { W[wo + 16], W[wo + (unsigned)ldw + 16] };
    v2f b2 = { W[wo + 32], W[wo + (unsigned)ldw + 32] };
    v2f b3 = { W[wo + 48], W[wo + (unsigned)ldw + 48] };
    c00 = wmma_f32(a0, b0, c00);  c10 = wmma_f32(a1, b0, c10);
    c01 = wmma_f32(a0, b1, c01);  c11 = wmma_f32(a1, b1, c11);
    c02 = wmma_f32(a0, b2, c02);  c12 = wmma_f32(a1, b2, c12);
    c03 = wmma_f32(a0, b3, c03);  c13 = wmma_f32(a1, b3, c13);
  }

  const float bi0 = bias[col0 +  0 + n];
  const float bi1 = bias[col0 + 16 + n];
  const float bi2 = bias[col0 + 32 + n];
  const float bi3 = bias[col0 + 48 + n];
  #pragma unroll
  for (int i = 0; i < 8; ++i) {
    const unsigned o0 = (unsigned)(row0 + i + hi8) * (unsigned)ldo
                      + (unsigned)(col0 + n);
    const unsigned o1 = o0 + 16u * (unsigned)ldo;
    float v00 = c00[i] + bi0, v01 = c01[i] + bi1;
    float v02 = c02[i] + bi2, v03 = c03[i] + bi3;
    float v10 = c10[i] + bi0, v11 = c11[i] + bi1;
    float v12 = c12[i] + bi2, v13 = c13[i] + bi3;
    if (relu) {
      v00 = fmaxf(v00, 0.f); v01 = fmaxf(v01, 0.f);
      v02 = fmaxf(v02, 0.f); v03 = fmaxf(v03, 0.f);
      v10 = fmaxf(v10, 0.f); v11 = fmaxf(v11, 0.f);
      v12 = fmaxf(v12, 0.f); v13 = fmaxf(v13, 0.f);
    }
    Out[o0] = v00; Out[o0 + 16] = v01; Out[o0 + 32] = v02; Out[o0 + 48] = v03;
    Out[o1] = v10; Out[o1 + 16] = v11; Out[o1 + 32] = v12; Out[o1 + 48] = v13;
  }
}

// ---------------------------------------------------------------------------
// Flash attention, WMMA everywhere. grid = (V/64, B*H), 4 waves/block.
// keys/vals layout: (H, B*V, D). Q = att[:, h*64 : h*64+64].
// Each wave: 16 query rows, online softmax over 64 key blocks of 16.
// S^T tile (M=w, N=q) keeps one query per lane -> cheap row stats.
// P is reshaped to A-fragment layout through a per-wave LDS tile.
// ---------------------------------------------------------------------------
__global__ __launch_bounds__(128) void attn_kernel(
    const float* __restrict__ att, const float* __restrict__ keys,
    const float* __restrict__ vals, float* __restrict__ out, float scale)
{
  const int bh = blockIdx.y;
  const int b = bh >> 3, h = bh & 7;
  const int lane = threadIdx.x & 31, wave = threadIdx.x >> 5;
  const int q0  = ((blockIdx.x << 2) | wave) << 4;
  const int n   = lane & 15;
  const int kb  = (lane >> 4) << 1;
  const int hi8 = (lane >> 4) << 3;
  __shared__ float pbuf[4][16][16];            // per-wave P^T tile

  // Preload Q fragments; layout doubles as B-fragment for S^T = K * Q^T.
  const unsigned qoff = (unsigned)(b * 1024 + q0 + n) * 512u
                      + (unsigned)(h * 64 + kb);
  v2f qa[16];
  #pragma unroll
  for (int kk = 0; kk < 16; ++kk) qa[kk] = *(const v2f*)(att + qoff + kk * 4);

  const unsigned kvbase = (unsigned)(h * 8192 + b * 1024) * 64u;
  v8f o0 = {}, o1 = {}, o2 = {}, o3 = {};
  float row_max = -3.0e38f, row_sum = 0.f;

  for (int j = 0; j < 64; ++j) {
    const int w0 = j << 4;
    // --- S^T(16w x 16q) = K_tile(16w x 64d) @ Q^T(64d x 16q), 16 WMMAs
    const unsigned koff = kvbase + (unsigned)(w0 + n) * 64u + (unsigned)kb;
    v8f s = {};
    #pragma unroll
    for (int kk = 0; kk < 16; ++kk) {
      v2f a = *(const v2f*)(keys + koff + kk * 4);
      s = wmma_f32(a, qa[kk], s);
    }
    // --- online softmax; lane owns column q=n (rows w=i+hi8 in-lane)
    float tmax = -3.0e38f;
    #pragma unroll
    for (int i = 0; i < 8; ++i) { s[i] *= scale; tmax = fmaxf(tmax, s[i]); }
    tmax = fmaxf(tmax, __shfl_xor(tmax, 16));
    const float nmax = fmaxf(row_max, tmax);
    const float corr = __expf(row_max - nmax);
    float tsum = 0.f;
    #pragma unroll
    for (int i = 0; i < 8; ++i) {
      const float pe = __expf(s[i] - nmax);
      pbuf[wave][i + hi8][n] = pe;           // P^T[w][q]
      tsum += pe;
    }
    tsum += __shfl_xor(tsum, 16);
    row_sum = row_sum * corr + tsum;
    row_max = nmax;
    // Rescale O accumulators: factor for row m=i+hi8 lives in lane (i+hi8).
    float f[8];
    #pragma unroll
    for (int i = 0; i < 8; ++i) f[i] = __shfl(corr, i + hi8);
    #pragma unroll
    for (int i = 0; i < 8; ++i) {
      o0[i] *= f[i]; o1[i] *= f[i]; o2[i] *= f[i]; o3[i] *= f[i];
    }
    // --- O(16q x 64d) += P(16q x 16w) @ V_tile(16w x 64d), 16 WMMAs
    #pragma unroll
    for (int kk = 0; kk < 4; ++kk) {
      v2f pa = { pbuf[wave][kk * 4 + kb][n], pbuf[wave][kk * 4 + kb + 1][n] };
      const unsigned voff = kvbase + (unsigned)(w0 + kk * 4 + kb) * 64u
                          + (unsigned)n;
      v2f b0 = { vals[voff +  0], vals[voff + 64 +  0] };
      v2f b1 = { vals[voff + 16], vals[voff + 64 + 16] };
      v2f b2 = { vals[voff + 32], vals[voff + 64 + 32] };
      v2f b3 = { vals[voff + 48], vals[voff + 64 + 48] };
      o0 = wmma_f32(pa, b0, o0);
      o1 = wmma_f32(pa, b1, o1);
      o2 = wmma_f32(pa, b2, o2);
      o3 = wmma_f32(pa, b3, o3);
    }
  }
  float rs[8];
  #pragma unroll
  for (int i = 0; i < 8; ++i) rs[i] = 1.f / __shfl(row_sum, i + hi8);
  #pragma unroll
  for (int i = 0; i < 8; ++i) {
    const unsigned oo = (unsigned)(b * 1024 + q0 + i + hi8) * 512u
                      + (unsigned)(h * 64 + n);
    out[oo +  0] = o0[i] * rs[i];
    out[oo + 16] = o1[i] * rs[i];
    out[oo + 32] = o2[i] * rs[i];
    out[oo + 48] = o3[i] * rs[i];
  }
}

// ---------------------------------------------------------------------------
// att[r] = LayerNorm(att[r] + delta[r]) * g + b   (row length 512)
// ---------------------------------------------------------------------------
__global__ __launch_bounds__(128) void add_ln_kernel(
    float* __restrict__ att, const float* __restrict__ delta,
    const float* __restrict__ g, const float* __restrict__ bta)
{
  const int r = blockIdx.x, t = threadIdx.x;
  const int lane = t & 31, wave = t >> 5;
  float4 x = ((const float4*)(att   + (unsigned)r * 512u))[t];
  float4 d = ((const float4*)(delta + (unsigned)r * 512u))[t];
  x.x += d.x; x.y += d.y; x.z += d.z; x.w += d.w;
  float s  = x.x + x.y + x.z + x.w;
  float sq = x.x * x.x + x.y * x.y + x.z * x.z + x.w * x.w;
  #pragma unroll
  for (int o = 16; o; o >>= 1) { s += __shfl_xor(s, o); sq += __shfl_xor(sq, o); }
  __shared__ float rs[4], rq[4];
  if (lane == 0) { rs[wave] = s; rq[wave] = sq; }
  __syncthreads();
  s  = rs[0] + rs[1] + rs[2] + rs[3];
  sq = rq[0] + rq[1] + rq[2] + rq[3];
  const float mean = s * (1.f / 512.f);
  const float var  = sq * (1.f / 512.f) - mean * mean;
  const float rstd = rsqrtf(var + 1e-5f);
  const float4 gg = ((const float4*)g)[t];
  const float4 bb = ((const float4*)bta)[t];
  float4 y;
  y.x = (x.x - mean) * rstd * gg.x + bb.x;
  y.y = (x.y - mean) * rstd * gg.y + bb.y;
  y.z = (x.z - mean) * rstd * gg.z + bb.z;
  y.w = (x.w - mean) * rstd * gg.w + bb.w;
  ((float4*)(att + (unsigned)r * 512u))[t] = y;
}

// ---------------------------------------------------------------------------
// u[r] = softmax(att[r] @ W_dist + b_dist) . centers   (512 -> 128)
// ---------------------------------------------------------------------------
__global__ __launch_bounds__(128) void dist_u_kernel(
    const float* __restrict__ att, const float* __restrict__ Wd,
    const float* __restrict__ bd, float* __restrict__ u)
{
  const int r = blockIdx.x, j = threadIdx.x;
  const int lane = j & 31, wave = j >> 5;
  __shared__ float xs[512];
  __shared__ float red[4];
  #pragma unroll
  for (int c = 0; c < 4; ++c)
    xs[c * 128 + j] = att[(unsigned)r * 512u + (unsigned)(c * 128 + j)];
  __syncthreads();
  float acc = bd[j];
  for (int k = 0; k < 512; ++k) acc = fmaf(xs[k], Wd[k * 128 + j], acc);
  // block max
  float m = acc;
  #pragma unroll
  for (int o = 16; o; o >>= 1) m = fmaxf(m, __shfl_xor(m, o));
  if (lane == 0) red[wave] = m;
  __syncthreads();
  m = fmaxf(fmaxf(red[0], red[1]), fmaxf(red[2], red[3]));
  __syncthreads();
  const float e = __expf(acc - m);
  float se = e;
  #pragma unroll
  for (int o = 16; o; o >>= 1) se += __shfl_xor(se, o);
  if (lane == 0) red[wave] = se;
  __syncthreads();
  se = red[0] + red[1] + red[2] + red[3];
  __syncthreads();
  float sc = e * ((j + 0.5f) * (1.0f / 128.0f));
  #pragma unroll
  for (int o = 16; o; o >>= 1) sc += __shfl_xor(sc, o);
  if (lane == 0) red[wave] = sc;
  __syncthreads();
  if (j == 0) u[r] = (red[0] + red[1] + red[2] + red[3]) / se;
}

// ---------------------------------------------------------------------------
// Deep sigmoidal flow: one thread per (b,v). params row = 3 * (16a,16b,16w).
// ---------------------------------------------------------------------------
__global__ __launch_bounds__(128) void dsf_kernel(
    const float* __restrict__ u, const float* __restrict__ params,
    float* __restrict__ out, int nrows)
{
  const int r = blockIdx.x * 128 + threadIdx.x;
  if (r >= nrows) return;
  float x = u[r], logdet = 0.f;
  const float* p = params + (unsigned)r * 144u;
  #pragma unroll
  for (int i = 0; i < 3; ++i, p += 48) {
    const float xc = fminf(fmaxf(x, -100.f), 100.f);
    float pw[16];
    float wmax = -3.0e38f;
    #pragma unroll
    for (int k = 0; k < 16; ++k) { pw[k] = p[32 + k]; wmax = fmaxf(wmax, pw[k]); }
    float wsum = 0.f;
    #pragma unroll
    for (int k = 0; k < 16; ++k) { pw[k] = __expf(pw[k] - wmax); wsum += pw[k]; }
    const float winv = 1.f / wsum;
    float xpre = 0.f, lj = 0.f;
    #pragma unroll
    for (int k = 0; k < 16; ++k) {
      const float pa = p[k];
      const float a = (pa > 0.f ? pa + log1pf(__expf(-pa)) : log1pf(__expf(pa))) + EPSF;
      const float z = a * xc + p[16 + k];
      float sg = 1.f / (1.f + __expf(-z));
      sg = fminf(fmaxf(sg, EPSF), 1.f - EPSF);
      const float w = pw[k] * winv;
      xpre += w * sg;
      lj   += w * sg * (1.f - sg) * a;
    }
    lj = fmaxf(lj, EPSF);
    logdet += logf(lj);
    if (i < 2) {
      const float xp  = fminf(fmaxf(xpre, EPSF), 1.f - EPSF);
      const float lxp = logf(xp), l1 = logf(1.f - xp);
      x = lxp - l1;
      logdet -= lxp + l1;
    } else {
      x = xpre;
    }
  }
  out[2 * r]     = x;
  out[2 * r + 1] = logdet;
}

// ---------------------------------------------------------------------------
extern "C" void kernel_launch(void* const* d_in, const int* in_sizes, int n_in,
                              void* d_out, int out_size, void* d_ws, size_t ws_size,
                              hipStream_t stream) {
  (void)in_sizes; (void)n_in; (void)out_size; (void)ws_size;
  constexpr int Bv = 8, V = 1024, E = 256, L = 4, H = 8, D = 64;
  constexpr int HD = 512, Mrows = Bv * V, FF = 1024;

  const float* X    = (const float*)d_in[0];
  const float* Wsh  = (const float*)d_in[1];
  const float* bsh  = (const float*)d_in[2];
  const float* Wk   = (const float*)d_in[3];
  const float* bk   = (const float*)d_in[4];
  const float* Wv   = (const float*)d_in[5];
  const float* bv   = (const float*)d_in[6];
  const float* ln1g = (const float*)d_in[7];
  const float* ln1b = (const float*)d_in[8];
  const float* ln2g = (const float*)d_in[9];
  const float* ln2b = (const float*)d_in[10];
  const float* Wff1 = (const float*)d_in[11];
  const float* bff1 = (const float*)d_in[12];
  const float* Wffm = (const float*)d_in[13];
  const float* bffm = (const float*)d_in[14];
  const float* Wff2 = (const float*)d_in[15];
  const float* bff2 = (const float*)d_in[16];
  const float* Wd   = (const float*)d_in[17];
  const float* bd   = (const float*)d_in[18];
  const float* dsfp = (const float*)d_in[19];

  float* ws       = (float*)d_ws;
  float* att      = ws;                                   // Mrows*HD
  float* attn_out = att      + (size_t)Mrows * HD;        // Mrows*HD
  float* keysbuf  = attn_out + (size_t)Mrows * HD;        // H*Mrows*D
  float* valsbuf  = keysbuf  + (size_t)H * Mrows * D;     // H*Mrows*D
  float* ff1      = valsbuf  + (size_t)H * Mrows * D;     // Mrows*FF
  float* ffm      = ff1      + (size_t)Mrows * FF;        // Mrows*FF
  float* ubuf     = ffm      + (size_t)Mrows * FF;        // Mrows

  const dim3 blk(128);

  // att = X @ W_shift + b_shift
  wmma_gemm_kernel<<<dim3(HD / 64, Mrows / 128, 1), blk, 0, stream>>>(
      X, E, Wsh, HD, 0, bsh, 0, att, HD, 0, E, 0);

  for (int l = 0; l < L; ++l) {
    // per-head K/V projections, batched over gridDim.z = H
    wmma_gemm_kernel<<<dim3(1, Mrows / 128, H), blk, 0, stream>>>(
        X, E, Wk + (size_t)l * H * E * D, D, E * D,
        bk + l * H * D, D, keysbuf, D, Mrows * D, E, 0);
    wmma_gemm_kernel<<<dim3(1, Mrows / 128, H), blk, 0, stream>>>(
        X, E, Wv + (size_t)l * H * E * D, D, E * D,
        bv + l * H * D, D, valsbuf, D, Mrows * D, E, 0);

    attn_kernel<<<dim3(V / 64, Bv * H), blk, 0, stream>>>(
        att, keysbuf, valsbuf, attn_out, 0.125f);

    add_ln_kernel<<<Mrows, 128, 0, stream>>>(att, attn_out,
                                             ln1g + l * HD, ln1b + l * HD);

    wmma_gemm_kernel<<<dim3(FF / 64, Mrows / 128, 1), blk, 0, stream>>>(
        att, HD, Wff1 + (size_t)l * HD * FF, FF, 0,
        bff1 + l * FF, 0, ff1, FF, 0, HD, 1);
    wmma_gemm_kernel<<<dim3(FF / 64, Mrows / 128, 1), blk, 0, stream>>>(
        ff1, FF, Wffm + (size_t)l * FF * FF, FF, 0,
        bffm + l * FF, 0, ffm, FF, 0, FF, 1);
    wmma_gemm_kernel<<<dim3(HD / 64, Mrows / 128, 1), blk, 0, stream>>>(
        ffm, FF, Wff2 + (size_t)l * FF * HD, HD, 0,
        bff2 + l * HD, 0, attn_out, HD, 0, FF, 0);

    add_ln_kernel<<<Mrows, 128, 0, stream>>>(att, attn_out,
                                             ln2g + l * HD, ln2b + l * HD);
  }

  dist_u_kernel<<<Mrows, 128, 0, stream>>>(att, Wd, bd, ubuf);
  dsf_kernel<<<Mrows / 128, 128, 0, stream>>>(ubuf, dsfp, (float*)d_out, Mrows);
}